// PINN_89455578841726
// MI455X (gfx1250) — compile-verified
//
#include <hip/hip_runtime.h>
#include <hip/hip_bf16.h>

// ---------------------------------------------------------------------------
// PINN jet-propagation kernel for MI455X (gfx1250, wave32).
//  - degree-4 Taylor jet through 1->8->8->2 tanh MLP (== nested jacfwd)
//  - layer-1 jet GEMM (5 orders of W1[8x8] @ h) on V_WMMA_F32_16X16X4_F32
//  - half-tile merge folded into the WMMA C operand (no readback shuffles)
//  - tanh on the TRANS unit via __builtin_amdgcn_tanhf when available
// ---------------------------------------------------------------------------

typedef float v2f __attribute__((ext_vector_type(2)));
typedef float v8f __attribute__((ext_vector_type(8)));

#if __has_builtin(__builtin_amdgcn_tanhf)
__device__ __forceinline__ float fast_tanh(float v) { return __builtin_amdgcn_tanhf(v); }
#else
__device__ __forceinline__ float fast_tanh(float v) { return tanhf(v); }
#endif

__global__ __launch_bounds__(128) void pinn_jet_kernel(
    const float* __restrict__ x,
    const float* __restrict__ W0, const float* __restrict__ b0,
    const float* __restrict__ W1, const float* __restrict__ b1,
    const float* __restrict__ W2, const float* __restrict__ b2,
    float* __restrict__ out, int n)
{
    const int tid  = blockIdx.x * blockDim.x + threadIdx.x;
    const int lane = threadIdx.x & 31;
    const int p    = tid;

    // Clamped load keeps every lane in-bounds -> EXEC stays all-1s for WMMA.
    const int idx = (p < n) ? p : (n - 1);

    // ---- layer 0: z = W0*(x/L0)+b0 ; jet of input is (x, 1/L0, 0, 0, 0) ----
    const float inv_L0 = 1.0f;                    // L0 == 1.0
    float xv = x[idx];
    float h  = xv * inv_L0;
    float h1 = inv_L0;

    float y[5][8];   // layer-0 activation jet: y[order][neuron], lane = point
    #pragma unroll
    for (int m = 0; m < 8; ++m) {
        float w  = W0[m];
        float z  = fmaf(w, h, b0[m]);
        float z1 = w * h1;                        // z2..z4 == 0 for layer 0
        float t  = fast_tanh(z);
        float s  = fmaf(-t, t, 1.0f);             // sech^2
        float f2 = -2.0f * t * s;
        float f3 = s * fmaf(6.0f * t, t, -2.0f);
        float f4 = t * s * fmaf(-24.0f * t, t, 16.0f);
        float z1_2 = z1 * z1;
        y[0][m] = t;
        y[1][m] = s  * z1;
        y[2][m] = f2 * z1_2;
        y[3][m] = f3 * z1_2 * z1;
        y[4][m] = f4 * z1_2 * z1_2;
    }

    // ---- layer 1 matmul (5 jet orders): z^(k) = W1 @ y^(k) on WMMA ----
    // A tile (constant per wave): A[m][k] = W1[m][k] for m < 8, else 0.
    // ISA 16x4 f32 A layout: VGPR0 = K{0|2}, VGPR1 = K{1|3};
    // lanes 0-15 -> M = lane, lanes 16-31 -> M = lane-16.
    const int arow = lane & 15;
    const int khi  = (lane >> 4) * 2;             // 0 or 2
    v2f A1, A2;
    A1.x = (arow < 8) ? W1[arow * 8 + khi + 0] : 0.0f;  // K = 0/2
    A1.y = (arow < 8) ? W1[arow * 8 + khi + 1] : 0.0f;  // K = 1/3
    A2.x = (arow < 8) ? W1[arow * 8 + khi + 4] : 0.0f;  // K = 4/6
    A2.y = (arow < 8) ? W1[arow * 8 + khi + 5] : 0.0f;  // K = 5/7

    const bool lowHalf = (lane < 16);

    // Batch A: columns = points of lanes 16-31.  B 4x16 layout:
    // VGPR0 = K{0|2}, VGPR1 = K{1|3}; col = lane&15.
    // Dest lanes 0-15 need y[K=0,1,4,5] from lane+16 (xor-swap);
    // dest lanes 16-31 need y[K=2,3,6,7] from their own lane.
    v8f SW[5];
    #pragma unroll
    for (int k = 0; k < 5; ++k) {
        float s0 = __shfl_xor(y[k][0], 16, 32);
        float s1 = __shfl_xor(y[k][1], 16, 32);
        float s4 = __shfl_xor(y[k][4], 16, 32);
        float s5 = __shfl_xor(y[k][5], 16, 32);
        v2f B1, B2;
        B1.x = lowHalf ? s0 : y[k][2];
        B1.y = lowHalf ? s1 : y[k][3];
        B2.x = lowHalf ? s4 : y[k][6];
        B2.y = lowHalf ? s5 : y[k][7];
        v8f c = {};
        // 8 args: (neg_a, A, neg_b, B, c_mod, C, reuse_a, reuse_b)
        c = __builtin_amdgcn_wmma_f32_16x16x4_f32(
                false, A1, false, B1, (short)0, c, false, false);
        c = __builtin_amdgcn_wmma_f32_16x16x4_f32(
                false, A2, false, B2, (short)0, c, false, false);
        // Rows 8-15 of c are exactly 0 (A rows 8-15 are 0, C = 0).
        // Swap halves: lanes 16-31 now hold batch-A z, lanes 0-15 hold 0.
        #pragma unroll
        for (int m = 0; m < 8; ++m)
            SW[k][m] = __shfl_xor(c[m], 16, 32);
    }

    // Batch B: columns = points of lanes 0-15, with C = SW so the final D is
    // already in elementwise layout: lane l < 16 -> rows 0-7 = batch-B z,
    // lane l >= 16 -> rows 8-15 = 0 + SW = batch-A z.
    float zjet[5][8];
    #pragma unroll
    for (int k = 0; k < 5; ++k) {
        float s2 = __shfl_xor(y[k][2], 16, 32);
        float s3 = __shfl_xor(y[k][3], 16, 32);
        float s6 = __shfl_xor(y[k][6], 16, 32);
        float s7 = __shfl_xor(y[k][7], 16, 32);
        v2f B1, B2;
        B1.x = lowHalf ? y[k][0] : s2;
        B1.y = lowHalf ? y[k][1] : s3;
        B2.x = lowHalf ? y[k][4] : s6;
        B2.y = lowHalf ? y[k][5] : s7;
        v8f c = SW[k];
        c = __builtin_amdgcn_wmma_f32_16x16x4_f32(
                false, A1, false, B1, (short)0, c, false, false);
        c = __builtin_amdgcn_wmma_f32_16x16x4_f32(
                false, A2, false, B2, (short)0, c, false, false);
        #pragma unroll
        for (int m = 0; m < 8; ++m)
            zjet[k][m] = c[m];
    }
    #pragma unroll
    for (int m = 0; m < 8; ++m) zjet[0][m] += b1[m];   // bias, order 0 only

    // ---- layer 1 tanh jet (full Faa di Bruno to 4th order) ----
    float g[5][8];
    #pragma unroll
    for (int m = 0; m < 8; ++m) {
        float t  = fast_tanh(zjet[0][m]);
        float s  = fmaf(-t, t, 1.0f);
        float f2 = -2.0f * t * s;
        float f3 = s * fmaf(6.0f * t, t, -2.0f);
        float f4 = t * s * fmaf(-24.0f * t, t, 16.0f);
        float z1 = zjet[1][m], z2 = zjet[2][m], z3 = zjet[3][m], z4 = zjet[4][m];
        float z1_2 = z1 * z1;
        g[0][m] = t;
        g[1][m] = s * z1;
        g[2][m] = fmaf(s, z2, f2 * z1_2);
        g[3][m] = fmaf(s, z3, fmaf(3.0f * f2, z1 * z2, f3 * z1_2 * z1));
        float t2 = f2 * fmaf(4.0f, z1 * z3, 3.0f * z2 * z2);
        float t3 = 6.0f * f3 * z1_2 * z2;
        float t4 = f4 * z1_2 * z1_2;
        g[4][m] = fmaf(s, z4, t2 + t3 + t4);
    }

    // ---- layer 2: o^(k) = W2 @ g^(k) (+b2 at order 0) ----
    float u0 = b2[0], u1 = 0.0f, u2 = 0.0f;
    float w0 = b2[1], w1 = 0.0f, w2 = 0.0f, w3 = 0.0f, w4 = 0.0f;
    #pragma unroll
    for (int q = 0; q < 8; ++q) {
        float wu = W2[q];          // row 0 of (2,8)
        float ww = W2[8 + q];      // row 1
        u0 = fmaf(wu, g[0][q], u0);
        u1 = fmaf(wu, g[1][q], u1);
        u2 = fmaf(wu, g[2][q], u2);
        w0 = fmaf(ww, g[0][q], w0);
        w1 = fmaf(ww, g[1][q], w1);
        w2 = fmaf(ww, g[2][q], w2);
        w3 = fmaf(ww, g[3][q], w3);
        w4 = fmaf(ww, g[4][q], w4);
    }

    if (p < n) {
        float4* dst = (float4*)(out + (size_t)p * 8);
        dst[0] = make_float4(u0, u1, u2, w0);
        dst[1] = make_float4(w1, w2, w3, w4);
    }
}

extern "C" void kernel_launch(void* const* d_in, const int* in_sizes, int n_in,
                              void* d_out, int out_size, void* d_ws, size_t ws_size,
                              hipStream_t stream) {
    const float* x  = (const float*)d_in[0];
    const float* W0 = (const float*)d_in[1];
    const float* b0 = (const float*)d_in[2];
    const float* W1 = (const float*)d_in[3];
    const float* b1 = (const float*)d_in[4];
    const float* W2 = (const float*)d_in[5];
    const float* b2 = (const float*)d_in[6];
    float* out = (float*)d_out;
    const int n = in_sizes[0];                 // N (x is (N,1))
    const int block = 128;                     // 4 waves; 2M % 128 == 0
    const int grid  = (n + block - 1) / block;
    hipLaunchKernelGGL(pinn_jet_kernel, dim3(grid), dim3(block), 0, stream,
                       x, W0, b0, W1, b1, W2, b2, out, n);
}